// VolumeRenderer_8615704396174
// MI455X (gfx1250) — compile-verified
//
#include <hip/hip_runtime.h>
#include <cstdint>

// Volume rendering (NeRF-style ray marching) for MI455X / gfx1250.
// Bandwidth-bound: ~320 MB in / 2 MB out -> ~14 us floor at 23.3 TB/s.
// Strategy: one wave32 per ray, async global->LDS double buffering (CDNA5
// ASYNCcnt path, non-temporal since data is consumed exactly once),
// wave-level multiply-scan for the exclusive transmittance cumprod,
// shuffle reduction for the 4 outputs.

#define EPS_T      1e-10f
#define FAR_DELTA  1e10f
#define P_SAMPLES  128
#define RPW        8      // rays processed per wave (pipeline depth amortizer)
#define WPB        8      // waves per block (256 threads)
#define RAY_BYTES  2560   // 512 density + 512 depth + 1536 feature

// Wait until ASYNCcnt <= n, then a compiler-level memory fence so no LDS
// load of the staged buffer can be scheduled above the wait (the builtin
// alone is not an IR memory barrier).
#if __has_builtin(__builtin_amdgcn_s_wait_asynccnt)
#define WAIT_ASYNC(n)                              \
  do {                                             \
    __builtin_amdgcn_s_wait_asynccnt(n);           \
    asm volatile("" ::: "memory");                 \
  } while (0)
#else
#define WAIT_ASYNC(n) asm volatile("s_wait_asynccnt " #n ::: "memory")
#endif

// One async DMA: 16 bytes per lane, 512 B per wave, LDS dest addr in VGPR,
// 64-bit global addr in VGPR pair (GV mode, saddr = off). Non-temporal:
// pure streaming, no reuse, don't thrash L2.
__device__ __forceinline__ void async_copy_b128(unsigned lds_off,
                                                unsigned long long gaddr) {
  asm volatile("global_load_async_to_lds_b128 %0, %1, off th:TH_LOAD_NT"
               :: "v"(lds_off), "v"(gaddr)
               : "memory");
}

// Stage one ray's density/depth/feature rows into an LDS buffer (5 async ops).
__device__ __forceinline__ void issue_ray(const float* __restrict__ density,
                                          const float* __restrict__ depth,
                                          const float* __restrict__ feature,
                                          long long ray, unsigned char* buf,
                                          int lane) {
  // Generic pointer to __shared__: low 32 bits are the LDS byte address.
  unsigned lds0 = (unsigned)(uintptr_t)buf + 16u * (unsigned)lane;
  unsigned long long gden =
      (unsigned long long)(uintptr_t)(density + ray * P_SAMPLES) + 16ull * (unsigned)lane;
  unsigned long long gdep =
      (unsigned long long)(uintptr_t)(depth + ray * P_SAMPLES) + 16ull * (unsigned)lane;
  unsigned long long gfea =
      (unsigned long long)(uintptr_t)(feature + ray * (P_SAMPLES * 3)) + 16ull * (unsigned)lane;
  async_copy_b128(lds0,          gden);            // density: 512 B
  async_copy_b128(lds0 +  512u,  gdep);            // depth:   512 B
  async_copy_b128(lds0 + 1024u,  gfea);            // feature: 1536 B
  async_copy_b128(lds0 + 1536u,  gfea +  512ull);
  async_copy_b128(lds0 + 2048u,  gfea + 1024ull);
}

__device__ __forceinline__ void compute_ray(const unsigned char* buf, int lane,
                                            long long ray, int N,
                                            float* __restrict__ out) {
  const float4* sv = (const float4*)(buf);            // density, lane l -> samples 4l..4l+3
  const float4* dv = (const float4*)(buf + 512);      // depth
  const float*  ds = (const float*)(buf + 512);       // depth, scalar view
  const float4* fv = (const float4*)(buf + 1024);     // feature (12 floats / lane)

  float4 sig = sv[lane];
  float4 dep = dv[lane];
  int ni = 4 * lane + 4; if (ni > P_SAMPLES - 1) ni = P_SAMPLES - 1;
  float dnext = ds[ni];                               // depth of sample 4l+4
  float4 f0 = fv[3 * lane + 0];
  float4 f1 = fv[3 * lane + 1];
  float4 f2 = fv[3 * lane + 2];

  // deltas (last global sample gets FAR_DELTA)
  float del0 = dep.y - dep.x;
  float del1 = dep.z - dep.y;
  float del2 = dep.w - dep.z;
  float del3 = (lane == 31) ? FAR_DELTA : (dnext - dep.w);

  // alpha = 1 - exp(-relu(sigma) * delta); t = 1 - alpha + eps
  float a0 = 1.0f - __expf(-fmaxf(sig.x, 0.0f) * del0);
  float a1 = 1.0f - __expf(-fmaxf(sig.y, 0.0f) * del1);
  float a2 = 1.0f - __expf(-fmaxf(sig.z, 0.0f) * del2);
  float a3 = 1.0f - __expf(-fmaxf(sig.w, 0.0f) * del3);
  float t0 = 1.0f - a0 + EPS_T;
  float t1 = 1.0f - a1 + EPS_T;
  float t2 = 1.0f - a2 + EPS_T;
  float t3 = 1.0f - a3 + EPS_T;

  // Exclusive cumprod across 128 samples: per-lane product, wave32 scan,
  // then within-lane expansion.
  float L = t0 * t1 * t2 * t3;
  float v = L;
#pragma unroll
  for (int d = 1; d < 32; d <<= 1) {
    float o = __shfl_up(v, d, 32);
    if (lane >= d) v *= o;
  }
  float excl = __shfl_up(v, 1, 32);   // product over lanes < l
  if (lane == 0) excl = 1.0f;

  float T0 = excl;
  float T1 = T0 * t0;
  float T2 = T1 * t1;
  float T3 = T2 * t2;

  float w0 = a0 * T0, w1 = a1 * T1, w2 = a2 * T2, w3 = a3 * T3;

  // Per-lane partial sums. Feature layout per lane: s0={f0.x,y,z},
  // s1={f0.w,f1.x,f1.y}, s2={f1.z,f1.w,f2.x}, s3={f2.y,f2.z,f2.w}.
  float sr = w0 * f0.x + w1 * f0.w + w2 * f1.z + w3 * f2.y;
  float sg = w0 * f0.y + w1 * f1.x + w2 * f1.w + w3 * f2.z;
  float sb = w0 * f0.z + w1 * f1.y + w2 * f2.x + w3 * f2.w;
  float sd = w0 * dep.x + w1 * dep.y + w2 * dep.z + w3 * dep.w;

#pragma unroll
  for (int m = 16; m > 0; m >>= 1) {
    sr += __shfl_xor(sr, m, 32);
    sg += __shfl_xor(sg, m, 32);
    sb += __shfl_xor(sb, m, 32);
    sd += __shfl_xor(sd, m, 32);
  }

  if (lane == 0) {
    out[ray * 3 + 0] = sr;
    out[ray * 3 + 1] = sg;
    out[ray * 3 + 2] = sb;
    out[(long long)N * 3 + ray] = sd;   // depth_out after feat_out
  }
}

__global__ __launch_bounds__(256)
void VolumeRenderer_gfx1250_kernel(const float* __restrict__ density,
                                   const float* __restrict__ feature,
                                   const float* __restrict__ depth,
                                   float* __restrict__ out, int N) {
  __shared__ alignas(16) unsigned char smem[WPB][2][RAY_BYTES];  // 40 KB/block

  const int lane = threadIdx.x & 31;
  const int wib  = threadIdx.x >> 5;
  const long long wave = (long long)blockIdx.x * WPB + wib;
  const long long base = wave * RPW;
  if (base >= N) return;

  unsigned char* b0 = smem[wib][0];
  unsigned char* b1 = smem[wib][1];

  // Prologue: stage first ray.
  issue_ray(density, depth, feature, base, b0, lane);

  for (int i = 0; i < RPW; ++i) {
    long long ray = base + i;
    if (ray >= N) break;                               // uniform per wave
    unsigned char* cur = (i & 1) ? b1 : b0;
    unsigned char* nxt = (i & 1) ? b0 : b1;
    bool more = (i + 1 < RPW) && (ray + 1 < N);
    if (more) {
      issue_ray(density, depth, feature, ray + 1, nxt, lane);  // prefetch next
      WAIT_ASYNC(5);   // loads complete in order: <=5 outstanding => cur done
    } else {
      WAIT_ASYNC(0);
    }
    compute_ray(cur, lane, ray, N, out);
  }
}

extern "C" void kernel_launch(void* const* d_in, const int* in_sizes, int n_in,
                              void* d_out, int out_size, void* d_ws, size_t ws_size,
                              hipStream_t stream) {
  (void)n_in; (void)d_ws; (void)ws_size; (void)out_size;
  const float* density = (const float*)d_in[0];   // [N, 128]
  const float* feature = (const float*)d_in[1];   // [N, 128, 3]
  const float* depth   = (const float*)d_in[2];   // [N, 128]
  float* out = (float*)d_out;                     // [N*3] feat ++ [N] depth

  const int N = in_sizes[0] / P_SAMPLES;
  const long long total_waves = ((long long)N + RPW - 1) / RPW;
  const int blocks = (int)((total_waves + WPB - 1) / WPB);

  VolumeRenderer_gfx1250_kernel<<<blocks, 256, 0, stream>>>(
      density, feature, depth, out, N);
}